// Pfaffian_21938692948133
// MI455X (gfx1250) — compile-verified
//
#include <hip/hip_runtime.h>
#include <hip/hip_bf16.h>

// Pfaffian of B=128 skew-symmetric 128x128 matrices via blocked Parlett-Reid.
// One workgroup (256 threads = 8 wave32) per batch element. Matrix lives in
// LDS (stride 129 -> conflict-free row AND column access). Panel of 16
// indices (8 pivot steps): strips updated with VALU, aggregated rank-16
// trailing update done with V_WMMA_F32_16X16X4_F32 (f32 WMMA; precision
// matters since the output is a product of pivots).

typedef float v2f __attribute__((ext_vector_type(2)));
typedef float v8f __attribute__((ext_vector_type(8)));

#define NPART 128
#define ASTR  129           // LDS row stride (129 mod 64 == 1: no bank conflicts)
#define PB    16            // panel width (8 pivot steps)
#define NTHREADS 256
#define SMEM_FLOATS (NPART * ASTR + NPART)   // A + idx
#define SMEM_BYTES  (SMEM_FLOATS * 4)

__global__ __launch_bounds__(NTHREADS)
void pfaffian_pr_wmma_kernel(const float* __restrict__ x,
                             const float* __restrict__ F,
                             float* __restrict__ out) {
  extern __shared__ float smem[];
  float* As = smem;                          // 128 x 129 f32
  int*   idx = (int*)(smem + NPART * ASTR);  // 128 ints

  const int b   = blockIdx.x;
  const int tid = threadIdx.x;

  // ---- 1. occupation indices: all j with x[j]>0, then all j with x[j]<=0 (+128)
  if (tid == 0) {
    const float* xb = x + b * NPART;
    int p = 0;
    for (int j = 0; j < NPART; ++j) if (xb[j] > 0.0f)    idx[p++] = j;
    for (int j = 0; j < NPART; ++j) if (!(xb[j] > 0.0f)) idx[p++] = j + NPART;
  }
  __syncthreads();

  // ---- 2. gather A[r][c] = F_full[idx[r], idx[c]]  (strict-lower packed F)
  for (int e = tid; e < NPART * NPART; e += NTHREADS) {
    const int r = e >> 7, c = e & 127;
    const int a = idx[r], q = idx[c];
    float v;
    if (a > q)      v =  F[((a * (a - 1)) >> 1) + q];
    else if (a < q) v = -F[((q * (q - 1)) >> 1) + a];
    else            v = 0.0f;
    As[r * ASTR + c] = v;
  }
  __syncthreads();

  const int lane = tid & 31;
  const int wave = tid >> 5;
  const int l15  = lane & 15;
  const int lhi  = lane >> 4;   // 0 or 1

  // ---- 3. blocked Parlett-Reid elimination
  for (int kp = 0; kp < NPART; kp += PB) {
    // 8 pivot steps inside the panel; keep panel row/col strips current
    for (int s = 0; s < PB / 2; ++s) {
      const int k = kp + 2 * s;

      // scale row k in place: tau[c] = A[k][c] / pivot for c > k+1
      if (tid < NPART) {
        const float pivot = As[k * ASTR + k + 1];
        const int c = tid;
        if (c > k + 1) As[k * ASTR + c] /= pivot;
      }
      __syncthreads();

      // column strip: c in [kp, kp+PB), r in (k+1, 128)
      for (int e = tid; e < NPART * PB; e += NTHREADS) {
        const int r = e >> 4;
        const int c = kp + (e & (PB - 1));
        if (r > k + 1 && c > k + 1) {
          const float tc = As[k * ASTR + c];       // tau[c]
          const float vr = As[r * ASTR + k + 1];   // col[r]
          const float vc = As[c * ASTR + k + 1];   // col[c]
          const float tr = As[k * ASTR + r];       // tau[r]
          As[r * ASTR + c] += tc * vr - vc * tr;
        }
      }
      // row strip: r in [kp, kp+PB) with r > k+1, c in [kp+PB, 128)
      for (int e = tid; e < PB * NPART; e += NTHREADS) {
        const int r = kp + (e >> 7);
        const int c = e & 127;
        if (r > k + 1 && c >= kp + PB) {
          const float tc = As[k * ASTR + c];
          const float vr = As[r * ASTR + k + 1];
          const float vc = As[c * ASTR + k + 1];
          const float tr = As[k * ASTR + r];
          As[r * ASTR + c] += tc * vr - vc * tr;
        }
      }
      __syncthreads();
    }

    // aggregated rank-16 trailing update with f32 WMMA:
    //   A_trail += U * V^T - V * U^T,
    //   U[:,s] = tau_s  = scaled row (kp+2s) of A,
    //   V[:,s] = col_s  = column (kp+2s+1) of A.
    const int t0 = kp + PB;
    const int nt = (NPART - t0) >> 4;       // trailing tiles per side
    if (nt > 0) {
      const int ntiles = nt * nt;
      for (int t = wave; t < ntiles; t += (NTHREADS / 32)) {  // wave-uniform
        const int rt = t0 + ((t / nt) << 4);
        const int ct = t0 + ((t % nt) << 4);

        // C tile (16x16 f32): VGPR g = rows rt+g (lanes 0-15) / rt+g+8 (16-31)
        v8f cfrag;
#pragma unroll
        for (int g = 0; g < 8; ++g)
          cfrag[g] = As[(rt + g + 8 * lhi) * ASTR + ct + l15];

#pragma unroll
        for (int kk = 0; kk < 8; kk += 4) {     // K = 8 in two K=4 chunks
          const int s0 = kk + 2 * lhi;          // A/B frag: lane half selects K pair
          v2f a, bf, a2, b2;
#pragma unroll
          for (int j = 0; j < 2; ++j) {
            const int ks = kp + 2 * (s0 + j);   // pivot row of step s0+j
            a[j]  =  As[ks * ASTR + rt + l15];          // U  (tau rows)
            bf[j] =  As[(ct + l15) * ASTR + ks + 1];    // V^T (col columns)
            a2[j] =  As[(rt + l15) * ASTR + ks + 1];    // V
            b2[j] = -As[ks * ASTR + ct + l15];          // -U^T (f32 WMMA has no A/B neg)
          }
          // C += U * V^T
          cfrag = __builtin_amdgcn_wmma_f32_16x16x4_f32(
              false, a, false, bf, (short)0, cfrag, false, false);
          // C += V * (-U^T)
          cfrag = __builtin_amdgcn_wmma_f32_16x16x4_f32(
              false, a2, false, b2, (short)0, cfrag, false, false);
        }

#pragma unroll
        for (int g = 0; g < 8; ++g)
          As[(rt + g + 8 * lhi) * ASTR + ct + l15] = cfrag[g];
      }
    }
    __syncthreads();
  }

  // ---- 4. Pfaffian = product of pivots (pivot entries are never overwritten)
  if (tid == 0) {
    float pf = 1.0f;
    for (int i = 0; i < NPART / 2; ++i)
      pf *= As[(2 * i) * ASTR + (2 * i + 1)];
    out[b] = pf;
  }
}

extern "C" void kernel_launch(void* const* d_in, const int* in_sizes, int n_in,
                              void* d_out, int out_size, void* d_ws, size_t ws_size,
                              hipStream_t stream) {
  const float* x = (const float*)d_in[0];   // (128, 128) f32
  const float* F = (const float*)d_in[1];   // 256*255/2 f32
  float* out = (float*)d_out;               // (128,) f32

  // 66.5 KB dynamic LDS (> default 64 KB cap) -> raise the limit.
  // Host-side attribute set; not a stream op, so graph capture is unaffected.
  (void)hipFuncSetAttribute((const void*)pfaffian_pr_wmma_kernel,
                            hipFuncAttributeMaxDynamicSharedMemorySize,
                            SMEM_BYTES);

  pfaffian_pr_wmma_kernel<<<128, NTHREADS, SMEM_BYTES, stream>>>(x, F, out);
}